// TF_calibration_block_12395275616344
// MI455X (gfx1250) — compile-verified
//
#include <hip/hip_runtime.h>
#include <cstddef>

// ---------------- problem constants (from reference setup_inputs) -----------
#define LLEN 4      // s_len == t_len
#define BB   8      // batch
#define CC   64
#define TT   256
#define DD   16
#define CDIM 1024   // C*D
#define HH   512    // frame MLP hidden
#define FF   32     // freq MLP width
#define EPSV 1e-6f

typedef __attribute__((ext_vector_type(2))) float v2f;
typedef __attribute__((ext_vector_type(8))) float v8f;

static __device__ __forceinline__ v8f wmma4(v2f a, v2f b, v8f c) {
  // V_WMMA_F32_16X16X4_F32 : D = A(16x4) * B(4x16) + C(16x16), fp32
  return __builtin_amdgcn_wmma_f32_16x16x4_f32(false, a, false, b, (short)0, c,
                                               false, false);
}

// Async global->LDS copy, 16B per lane, tracked by ASYNCcnt (CDNA5 path).
// lds operand = workgroup-relative LDS byte offset (low 32 bits of generic ptr).
static __device__ __forceinline__ void async_b128(const float* g, unsigned lds) {
  asm volatile("global_load_async_to_lds_b128 %0, %1, off"
               :: "v"(lds), "v"(g)
               : "memory");
}
static __device__ __forceinline__ void wait_async0() {
#if __has_builtin(__builtin_amdgcn_s_wait_asynccnt)
  __builtin_amdgcn_s_wait_asynccnt(0);
#else
  asm volatile("s_wait_asynccnt 0x0" ::: "memory");
#endif
}
static __device__ __forceinline__ unsigned lds_off(const void* p) {
  return (unsigned)(size_t)p;
}

// ---------------------------------------------------------------------------
// 1) [L,B,C,T,D] -> [L,B,T,C*D]  (float4 granular copy; D=16 divisible by 4)
__global__ void permute_feat(const float* __restrict__ in, float* __restrict__ out) {
  int idx = blockIdx.x * blockDim.x + threadIdx.x;      // float4 index
  // idx = ((lb*CC + c)*TT + t)*4 + d4  == input float4 index
  int d4 = idx & 3;
  int t  = (idx >> 2) & 255;
  int c  = (idx >> 10) & 63;
  int lb = idx >> 16;                                   // l*BB + b  (0..31)
  const float4* pin = (const float4*)in;
  float4 v = pin[idx];
  float4* pout = (float4*)out;
  pout[(size_t)(lb * TT + t) * (CDIM / 4) + c * (DD / 4) + d4] = v;
}

// ---------------------------------------------------------------------------
// 2) G[l,b] = X * X^T, X = x[l,b] in [TT, CDIM]. WMMA f32 16x16x4.
//    Block 256 thr = 8 waves, 64x64 tile, double-buffered async K-chunks of 32.
__global__ void gram_time(const float* __restrict__ x, float* __restrict__ G) {
  __shared__ float lA[2][64][36];   // row pitch 144B (16B aligned), even pad
  __shared__ float lB[2][64][36];
  const unsigned ABUF = 64 * 36 * 4;     // byte stride between buffers
  int lb = blockIdx.y;
  int tile = blockIdx.x;                 // 16 tiles (4x4) of 64x64
  int r0 = (tile >> 2) * 64, c0 = (tile & 3) * 64;
  const float* X = x + (size_t)lb * TT * CDIM;
  int tid = threadIdx.x, lane = tid & 31, wave = tid >> 5;
  // wave's two 16x16 subtiles: rows ar0 / ar0+32, same cols ac0
  int ar0 = (wave >> 2) * 16, ac0 = (wave & 3) * 16;
  v8f acc0 = {0.f,0.f,0.f,0.f,0.f,0.f,0.f,0.f};
  v8f acc1 = {0.f,0.f,0.f,0.f,0.f,0.f,0.f,0.f};
  int mm = lane & 15;                    // M/N index within fragment
  int kk = (lane >> 4) * 2;              // K sub-offset: 0 or 2
  int lr = tid >> 3, lk = (tid & 7) * 4; // per-thread loader coords
  unsigned la0 = lds_off(&lA[0][lr][lk]);
  unsigned la1 = lds_off(&lA[0][lr + 32][lk]);
  unsigned lb0 = lds_off(&lB[0][lr][lk]);
  unsigned lb1 = lds_off(&lB[0][lr + 32][lk]);

#define GRAM_ISSUE(p, kb)                                                     \
  do {                                                                        \
    async_b128(&X[(size_t)(r0 + lr) * CDIM + (kb) + lk], la0 + (p) * ABUF);   \
    async_b128(&X[(size_t)(r0 + lr + 32) * CDIM + (kb) + lk], la1 + (p) * ABUF); \
    async_b128(&X[(size_t)(c0 + lr) * CDIM + (kb) + lk], lb0 + (p) * ABUF);   \
    async_b128(&X[(size_t)(c0 + lr + 32) * CDIM + (kb) + lk], lb1 + (p) * ABUF); \
  } while (0)

  GRAM_ISSUE(0, 0);
  const int NCHUNK = CDIM / 32;
  for (int i = 0; i < NCHUNK; i++) {
    int cur = i & 1;
    wait_async0();          // my copies for chunk i have landed in LDS
    __syncthreads();        // publish to all waves / retire reads of buf[cur^1]
    if (i + 1 < NCHUNK) GRAM_ISSUE(cur ^ 1, (i + 1) * 32);
    const float (*At)[36] = lA[cur];
    const float (*Bt)[36] = lB[cur];
#pragma unroll
    for (int k0 = 0; k0 < 32; k0 += 4) {
      v2f a0 = *(const v2f*)&At[ar0 + mm][k0 + kk];
      v2f a1 = *(const v2f*)&At[ar0 + 32 + mm][k0 + kk];
      v2f b  = *(const v2f*)&Bt[ac0 + mm][k0 + kk];   // B[k][n] = X[col][k]
      acc0 = wmma4(a0, b, acc0);
      acc1 = wmma4(a1, b, acc1);
    }
  }
#undef GRAM_ISSUE
  float* Gp = G + (size_t)lb * TT * TT;
  int hi = lane >> 4, nn = lane & 15;
#pragma unroll
  for (int r = 0; r < 8; r++) {
    Gp[(size_t)(r0 + ar0 + r + 8 * hi) * TT + c0 + ac0 + nn] = acc0[r];
    Gp[(size_t)(r0 + ar0 + 32 + r + 8 * hi) * TT + c0 + ac0 + nn] = acc1[r];
  }
}

// ---------------------------------------------------------------------------
// 3) Gg[l,b,t,a] = x[l,b,t,:] . x[l,a,t,:]  (8x8 gram per (l,t), K=1024)
__global__ void gram_freq(const float* __restrict__ x, float* __restrict__ Gg) {
  __shared__ float rows[BB][CDIM + 4];     // +4: float4 aligned + de-bank
  int lt = blockIdx.x;                     // l*TT + t
  int l = lt >> 8, t = lt & 255;
  const float* base = x + (size_t)l * BB * TT * CDIM + (size_t)t * CDIM;
  int tid = threadIdx.x;                   // 64 threads
  for (int i = tid; i < BB * (CDIM / 4); i += 64) {
    int b = i >> 8, k4 = i & 255;
    ((float4*)&rows[b][0])[k4] =
        *(const float4*)&base[(size_t)b * TT * CDIM + (size_t)k4 * 4];
  }
  __syncthreads();
  int b = tid >> 3, a = tid & 7;
  float acc = 0.f;
  for (int k = 0; k < CDIM; k++) acc += rows[b][k] * rows[a][k];
  Gg[((size_t)(l * BB + b) * TT + t) * BB + a] = acc;
}

// ---------------------------------------------------------------------------
// 4) diag norms:  dn[lb,t] = sqrt(G[lb,t,t]) + EPS   (and freq variant)
__global__ void diag_norm_time(const float* __restrict__ G, float* __restrict__ dn) {
  int idx = blockIdx.x * blockDim.x + threadIdx.x;      // (l*B+b)*T + t
  if (idx >= LLEN * BB * TT) return;
  int t = idx & 255;
  dn[idx] = sqrtf(G[(size_t)idx * TT + t]) + EPSV;
}
__global__ void diag_norm_freq(const float* __restrict__ Gg, float* __restrict__ dn) {
  int idx = blockIdx.x * blockDim.x + threadIdx.x;
  if (idx >= LLEN * BB * TT) return;
  int b = (idx >> 8) & 7;
  dn[idx] = sqrtf(Gg[(size_t)idx * BB + b]) + EPSV;
}

// ---------------------------------------------------------------------------
// 5) Generic batched fp32 WMMA GEMM:  C[l] = act(A[l] @ B[l] + bias[l])
//    A [M,K] row-major, B [K,N] row-major. Tile 64x64, double-buffered Kc=32.
__global__ void gemm_bias(const float* __restrict__ A, const float* __restrict__ Bm,
                          const float* __restrict__ bias, float* __restrict__ C,
                          int M, int N, int K,
                          int strideA, int strideB, int strideBias, int strideC,
                          int relu) {
  __shared__ float lA[2][64][36];   // 144B row pitch
  __shared__ float lB[2][32][68];   // 272B row pitch
  const unsigned ABUF = 64 * 36 * 4;
  const unsigned BBUF = 32 * 68 * 4;
  int l = blockIdx.z;
  const float* Ap = A + (size_t)l * strideA;
  const float* Bp = Bm + (size_t)l * strideB;
  const float* bp = bias + (size_t)l * strideBias;
  float* Cp = C + (size_t)l * strideC;
  int m0 = blockIdx.y * 64, n0 = blockIdx.x * 64;
  int tid = threadIdx.x, lane = tid & 31, wave = tid >> 5;
  int ar0 = (wave >> 2) * 16, ac0 = (wave & 3) * 16;   // subtiles: ar0 / ar0+32
  v8f acc0 = {0.f,0.f,0.f,0.f,0.f,0.f,0.f,0.f};
  v8f acc1 = {0.f,0.f,0.f,0.f,0.f,0.f,0.f,0.f};
  int mm = lane & 15, kk = (lane >> 4) * 2;
  int lr = tid >> 3, lk = (tid & 7) * 4;     // A loader: 64 rows x 32 k
  int br = tid >> 4, bc = (tid & 15) * 4;    // B loader: 32 rows x 64 n
  unsigned la0 = lds_off(&lA[0][lr][lk]);
  unsigned la1 = lds_off(&lA[0][lr + 32][lk]);
  unsigned lb0 = lds_off(&lB[0][br][bc]);
  unsigned lb1 = lds_off(&lB[0][br + 16][bc]);

#define GEMM_ISSUE(p, kb)                                                        \
  do {                                                                           \
    async_b128(&Ap[(size_t)(m0 + lr) * K + (kb) + lk], la0 + (p) * ABUF);        \
    async_b128(&Ap[(size_t)(m0 + lr + 32) * K + (kb) + lk], la1 + (p) * ABUF);   \
    async_b128(&Bp[(size_t)((kb) + br) * N + n0 + bc], lb0 + (p) * BBUF);        \
    async_b128(&Bp[(size_t)((kb) + br + 16) * N + n0 + bc], lb1 + (p) * BBUF);   \
  } while (0)

  GEMM_ISSUE(0, 0);
  const int NCHUNK = K / 32;
  for (int i = 0; i < NCHUNK; i++) {
    int cur = i & 1;
    wait_async0();
    __syncthreads();
    if (i + 1 < NCHUNK) GEMM_ISSUE(cur ^ 1, (i + 1) * 32);
    const float (*At)[36] = lA[cur];
    const float (*Bt)[68] = lB[cur];
#pragma unroll
    for (int k0 = 0; k0 < 32; k0 += 4) {
      v2f a0 = *(const v2f*)&At[ar0 + mm][k0 + kk];
      v2f a1 = *(const v2f*)&At[ar0 + 32 + mm][k0 + kk];
      v2f b;
      b.x = Bt[k0 + kk][ac0 + mm];
      b.y = Bt[k0 + kk + 1][ac0 + mm];
      acc0 = wmma4(a0, b, acc0);
      acc1 = wmma4(a1, b, acc1);
    }
  }
#undef GEMM_ISSUE
  int hi = lane >> 4, nn = lane & 15;
#pragma unroll
  for (int r = 0; r < 8; r++) {
    int row0 = m0 + ar0 + r + 8 * hi, col = n0 + ac0 + nn;
    float v0 = acc0[r] + bp[col];
    if (relu) v0 = fmaxf(v0, 0.f);
    Cp[(size_t)row0 * N + col] = v0;
    int row1 = row0 + 32;
    float v1 = acc1[r] + bp[col];
    if (relu) v1 = fmaxf(v1, 0.f);
    Cp[(size_t)row1 * N + col] = v1;
  }
}

// ---------------------------------------------------------------------------
// 6) L2-normalize over the T axis of [L,B,T,F] (divide by sqrt(sum_t v^2))
__global__ void colnorm(float* __restrict__ buf, int F, int total) {
  int idx = blockIdx.x * blockDim.x + threadIdx.x;   // (l*B+b)*F + f
  if (idx >= total) return;
  int f = idx % F, lb = idx / F;
  float* p = buf + (size_t)lb * TT * F + f;
  float s = 0.f;
  for (int t = 0; t < TT; t++) { float v = p[(size_t)t * F]; s += v * v; }
  float inv = 1.0f / sqrtf(s);
  for (int t = 0; t < TT; t++) p[(size_t)t * F] *= inv;
}

// ---------------------------------------------------------------------------
// 7) tiny freq MLP: [.,8] -> relu([.,32]) -> [.,32]  (per (l,b,t))
__global__ void freq_mlp(const float* __restrict__ Gg, const float* __restrict__ w1,
                         const float* __restrict__ b1, const float* __restrict__ w2,
                         const float* __restrict__ b2, float* __restrict__ out) {
  int idx = blockIdx.x * blockDim.x + threadIdx.x;   // (l*B+b)*T + t
  if (idx >= LLEN * BB * TT) return;
  int l = idx >> 11;
  const float* xin = Gg + (size_t)idx * BB;
  float xv[8];
#pragma unroll
  for (int i = 0; i < 8; i++) xv[i] = xin[i];
  const float* W1 = w1 + l * 8 * FF;  const float* B1 = b1 + l * FF;
  const float* W2 = w2 + l * FF * FF; const float* B2 = b2 + l * FF;
  float h[FF];
#pragma unroll
  for (int j = 0; j < FF; j++) {
    float a = B1[j];
#pragma unroll
    for (int i = 0; i < 8; i++) a += xv[i] * W1[i * FF + j];
    h[j] = fmaxf(a, 0.f);
  }
  float* op = out + (size_t)idx * FF;
  for (int j = 0; j < FF; j++) {
    float a = B2[j];
#pragma unroll
    for (int i = 0; i < FF; i++) a += h[i] * W2[i * FF + j];
    op[j] = a;
  }
}

// ---------------------------------------------------------------------------
// 8) attention: out[b,t,i,j] = softmax_j( q[i,b,t,:] . k[j,b,t,:] )
__global__ void attention(const float* __restrict__ q, const float* __restrict__ k,
                          float* __restrict__ out, int S) {
  int idx = blockIdx.x * blockDim.x + threadIdx.x;   // b*T + t
  if (idx >= BB * TT) return;
  float logit[4][4];
  for (int i = 0; i < 4; i++)
    for (int j = 0; j < 4; j++) {
      const float* qp = q + ((size_t)i * BB * TT + idx) * S;
      const float* kp = k + ((size_t)j * BB * TT + idx) * S;
      float acc = 0.f;
      for (int s4 = 0; s4 < S; s4 += 4) {
        float4 qa = *(const float4*)&qp[s4];
        float4 ka = *(const float4*)&kp[s4];
        acc += qa.x * ka.x + qa.y * ka.y + qa.z * ka.z + qa.w * ka.w;
      }
      logit[i][j] = acc;
    }
  for (int i = 0; i < 4; i++) {
    float mx = logit[i][0];
    for (int j = 1; j < 4; j++) mx = fmaxf(mx, logit[i][j]);
    float e[4], se = 0.f;
    for (int j = 0; j < 4; j++) { e[j] = __expf(logit[i][j] - mx); se += e[j]; }
    for (int j = 0; j < 4; j++) out[((size_t)idx * 4 + i) * 4 + j] = e[j] / se;
  }
}

// ---------------------------------------------------------------------------
// 9) fused cos-normalize + affine + row-normalize + 4-way broadcast (time).
//    One wave per row (l,b,t); writes [4,4,B,T,T] slab of d_out.
__global__ void time_cos_broadcast(const float* __restrict__ G,
                                   const float* __restrict__ dn,
                                   float* __restrict__ out, int is_tea) {
  int wave = threadIdx.x >> 5, lane = threadIdx.x & 31;
  int row = blockIdx.x * 8 + wave;      // (l*B+b)*T + t , total 8192
  const float* g = G + (size_t)row * TT;
  float dt = dn[row];
  int rowbase = row & ~255;             // (l*B+b)*T
  float vals[8];
  float s = 0.f;
#pragma unroll
  for (int i = 0; i < 8; i++) {
    int col = lane + i * 32;
    float dc = dn[rowbase + col];
    float v = (g[col] / (dt * dc) + 1.f) * 0.5f;
    vals[i] = v; s += v;
  }
  for (int off = 16; off; off >>= 1) s += __shfl_xor(s, off, 32);
  float inv = 1.f / s;
  int l = row >> 11;                    // layer index
  int bt = row & 2047;                  // b*T + t
#pragma unroll
  for (int i = 0; i < 8; i++) {
    int col = lane + i * 32;
    float v = vals[i] * inv;
    for (int j = 0; j < 4; j++) {
      int ij = is_tea ? (j * 4 + l) : (l * 4 + j);   // tea: [i][j]=tea[j]
      out[((size_t)ij * BB * TT + bt) * TT + col] = v;
    }
  }
}

// 10) same for freq:  [4,4,B,T,8] slabs
__global__ void freq_cos_broadcast(const float* __restrict__ Gg,
                                   const float* __restrict__ dn,
                                   float* __restrict__ out, int is_tea) {
  int idx = blockIdx.x * blockDim.x + threadIdx.x;   // (l*B+b)*T + t
  if (idx >= LLEN * BB * TT) return;
  int l = idx >> 11;
  int t = idx & 255, b = (idx >> 8) & 7;
  const float* g = Gg + (size_t)idx * BB;
  float db = dn[idx];
  float vals[8], s = 0.f;
#pragma unroll
  for (int a = 0; a < 8; a++) {
    float da = dn[((l * BB + a) << 8) + t];
    float v = (g[a] / (db * da) + 1.f) * 0.5f;
    vals[a] = v; s += v;
  }
  float inv = 1.f / s;
  int bt = (b << 8) + t;
#pragma unroll
  for (int a = 0; a < 8; a++) {
    float v = vals[a] * inv;
    for (int j = 0; j < 4; j++) {
      int ij = is_tea ? (j * 4 + l) : (l * 4 + j);
      out[((size_t)ij * BB * TT + bt) * BB + a] = v;
    }
  }
}

// ---------------------------------------------------------------------------
extern "C" void kernel_launch(void* const* d_in, const int* in_sizes, int n_in,
                              void* d_out, int out_size, void* d_ws, size_t ws_size,
                              hipStream_t stream) {
  (void)in_sizes; (void)n_in; (void)out_size; (void)ws_size;
  const float* feat_s = (const float*)d_in[0];
  const float* feat_t = (const float*)d_in[1];
  const float* tk_w1 = (const float*)d_in[2];  const float* tk_b1 = (const float*)d_in[3];
  const float* tk_w2 = (const float*)d_in[4];  const float* tk_b2 = (const float*)d_in[5];
  const float* fk_w1 = (const float*)d_in[6];  const float* fk_b1 = (const float*)d_in[7];
  const float* fk_w2 = (const float*)d_in[8];  const float* fk_b2 = (const float*)d_in[9];
  const float* tq_w1 = (const float*)d_in[10]; const float* tq_b1 = (const float*)d_in[11];
  const float* tq_w2 = (const float*)d_in[12]; const float* tq_b2 = (const float*)d_in[13];
  const float* fq_w1 = (const float*)d_in[14]; const float* fq_b1 = (const float*)d_in[15];
  const float* fq_w2 = (const float*)d_in[16]; const float* fq_b2 = (const float*)d_in[17];

  // workspace layout (floats); total ~30.05M floats ~= 120 MB
  float* ws   = (float*)d_ws;
  float* xs   = ws;                 // [4,8,256,1024]  8388608
  float* xt   = xs  + 8388608;      // 8388608
  float* Gs   = xt  + 8388608;      // [4,8,256,256]   2097152
  float* Gt   = Gs  + 2097152;
  float* Gsg  = Gt  + 2097152;      // [4,8,256,8]     65536
  float* Gtg  = Gsg + 65536;
  float* Hb   = Gtg + 65536;        // [4,2048,512]    4194304 (reused)
  float* tkb  = Hb  + 4194304;      // [4,8,256,256]   2097152
  float* tqb  = tkb + 2097152;
  float* fkb  = tqb + 2097152;      // [4,8,256,32]    262144
  float* fqb  = fkb + 262144;
  float* dns  = fqb + 262144;       // [4*8*256] diag norms
  float* dnt  = dns + 8192;
  float* dnsg = dnt + 8192;
  float* dntg = dnsg + 8192;

  float* out        = (float*)d_out;
  float* o_stu_time = out;                       // [4,4,8,256,256] 8388608
  float* o_stu_freq = out + 8388608;             // [4,4,8,256,8]   262144
  float* o_tea_time = o_stu_freq + 262144;       // 8388608
  float* o_tea_freq = o_tea_time + 8388608;      // 262144
  float* o_time_att = o_tea_freq + 262144;       // [8,256,4,4]     32768
  float* o_freq_att = o_time_att + 32768;        // 32768

  // 1) permute feats
  permute_feat<<<8192, 256, 0, stream>>>(feat_s, xs);
  permute_feat<<<8192, 256, 0, stream>>>(feat_t, xt);

  // 2) big grams (WMMA + async LDS) + small batch grams
  dim3 gg(16, LLEN * BB);
  gram_time<<<gg, 256, 0, stream>>>(xs, Gs);
  gram_time<<<gg, 256, 0, stream>>>(xt, Gt);
  gram_freq<<<LLEN * TT, 64, 0, stream>>>(xs, Gsg);
  gram_freq<<<LLEN * TT, 64, 0, stream>>>(xt, Gtg);

  // 3) diagonal norms for cosine reuse
  diag_norm_time<<<32, 256, 0, stream>>>(Gs, dns);
  diag_norm_time<<<32, 256, 0, stream>>>(Gt, dnt);
  diag_norm_freq<<<32, 256, 0, stream>>>(Gsg, dnsg);
  diag_norm_freq<<<32, 256, 0, stream>>>(Gtg, dntg);

  // 4) frame MLPs (WMMA GEMMs): tk from Gt, tq from Gs; Hb reused (in-order stream)
  gemm_bias<<<dim3(HH / 64, 2048 / 64, LLEN), 256, 0, stream>>>(
      Gt, tk_w1, tk_b1, Hb, 2048, HH, TT, BB * TT * TT, TT * HH, HH, 2048 * HH, 1);
  gemm_bias<<<dim3(TT / 64, 2048 / 64, LLEN), 256, 0, stream>>>(
      Hb, tk_w2, tk_b2, tkb, 2048, TT, HH, 2048 * HH, HH * TT, TT, BB * TT * TT, 0);
  colnorm<<<32, 256, 0, stream>>>(tkb, TT, LLEN * BB * TT);
  gemm_bias<<<dim3(HH / 64, 2048 / 64, LLEN), 256, 0, stream>>>(
      Gs, tq_w1, tq_b1, Hb, 2048, HH, TT, BB * TT * TT, TT * HH, HH, 2048 * HH, 1);
  gemm_bias<<<dim3(TT / 64, 2048 / 64, LLEN), 256, 0, stream>>>(
      Hb, tq_w2, tq_b2, tqb, 2048, TT, HH, 2048 * HH, HH * TT, TT, BB * TT * TT, 0);
  colnorm<<<32, 256, 0, stream>>>(tqb, TT, LLEN * BB * TT);

  // 5) freq MLPs
  freq_mlp<<<32, 256, 0, stream>>>(Gtg, fk_w1, fk_b1, fk_w2, fk_b2, fkb);
  colnorm<<<4, 256, 0, stream>>>(fkb, FF, LLEN * BB * FF);
  freq_mlp<<<32, 256, 0, stream>>>(Gsg, fq_w1, fq_b1, fq_w2, fq_b2, fqb);
  colnorm<<<4, 256, 0, stream>>>(fqb, FF, LLEN * BB * FF);

  // 6) attention outputs
  attention<<<8, 256, 0, stream>>>(tqb, tkb, o_time_att, TT);
  attention<<<8, 256, 0, stream>>>(fqb, fkb, o_freq_att, FF);

  // 7) cosine + row-normalize + broadcast outputs
  time_cos_broadcast<<<1024, 256, 0, stream>>>(Gs, dns, o_stu_time, 0);
  time_cos_broadcast<<<1024, 256, 0, stream>>>(Gt, dnt, o_tea_time, 1);
  freq_cos_broadcast<<<32, 256, 0, stream>>>(Gsg, dnsg, o_stu_freq, 0);
  freq_cos_broadcast<<<32, 256, 0, stream>>>(Gtg, dntg, o_tea_freq, 1);
}